// MarginKLDivLossColBERT_15152644621117
// MI455X (gfx1250) — compile-verified
//
#include <hip/hip_runtime.h>
#include <hip/hip_bf16.h>

// Problem constants (from setup_inputs)
#define B_    128
#define LQ    32
#define LD    512
#define DIM   128
#define NNEG  8
#define NDOC  9          // doc 0 = pos, docs 1..8 = neg 0..7
#define EPSN  1e-12f
#define NEG_INF (-3.402823466e+38f)

// LDS row stride in floats: 132 = 128 + 4 pad. Row stride mod 64 banks = 4,
// so the 16 rows of a fragment read land on distinct bank pairs, and the two
// lane halves (kb offset +2) cover the complementary banks -> conflict-free.
#define SLAB_LD 132

// Dynamic LDS layout (floats):
//   Btile  : 8 * 16 * SLAB_LD = 16896  (per-wave masked doc tiles, 66 KB)
//   Asm    : LQ * SLAB_LD     = 4224   (A = q_n * invnorm)
//   ssq    : 8 * DIM          = 1024   (per-wave column sum-of-squares)
//   invn   : DIM              = 128
//   rowmax : 8 * LQ           = 256
// total ~88 KB -> 3 blocks per 320 KB WGP -> phase-1 streaming of some blocks
// overlaps phase-2 WMMA of others.
#define SMEM_FLOATS (8 * 16 * SLAB_LD + LQ * SLAB_LD + 8 * DIM + DIM + 8 * LQ)

typedef float v2f __attribute__((ext_vector_type(2)));
typedef float v4f __attribute__((ext_vector_type(4)));
typedef float v8f __attribute__((ext_vector_type(8)));

// ---------------------------------------------------------------------------
// Kernel 1: q = l2norm(q_emb, axis=D). One block per batch element, 8 waves,
// each wave normalizes 4 query tokens (wave32: 32 lanes * float4 = 128 dims).
// ---------------------------------------------------------------------------
__global__ __launch_bounds__(256) void qnorm_kernel(const float* __restrict__ q_emb,
                                                    float* __restrict__ q_n) {
    const int b    = blockIdx.x;
    const int w    = threadIdx.x >> 5;
    const int lane = threadIdx.x & 31;
    #pragma unroll
    for (int j = 0; j < 4; ++j) {
        const int t = w * 4 + j;
        const float* src = q_emb + ((size_t)b * LQ + t) * DIM;
        float4 v = *(const float4*)(src + lane * 4);
        float ss = v.x * v.x + v.y * v.y + v.z * v.z + v.w * v.w;
        #pragma unroll
        for (int off = 1; off < 32; off <<= 1)
            ss += __shfl_xor(ss, off, 32);
        const float inv = 1.0f / fmaxf(sqrtf(ss), EPSN);
        float4 o = make_float4(v.x * inv, v.y * inv, v.z * inv, v.w * inv);
        *(float4*)(q_n + ((size_t)b * LQ + t) * DIM + lane * 4) = o;
    }
}

// ---------------------------------------------------------------------------
// Kernel 2 (fused, L2-mediated): one block per (doc, b).
// Phase 1: stream the doc slab once (RT hint -> stays in 192 MB L2), skipping
//          mask==0 token rows (uniform branch; ~50% of bytes never read),
//          accumulating token-axis sum-of-squares per dim in registers.
// Phase 2: invnorm[d] = 1/max(||col d||,eps) (the F.normalize(dim=1) quirk),
//          A = q_n[b] * invnorm, then per-wave 16-token B tiles re-read
//          through L2 with last-use hints into padded LDS, and the MaxSim
//          GEMM via V_WMMA_F32_16X16X4_F32 (8 waves x 4 N-tiles x 2 M-tiles
//          x 32 K-steps), running per-row max in registers.
// HBM traffic ~134 MB expected (masked rows skipped) -> ~6 us floor.
// ---------------------------------------------------------------------------
__global__ __launch_bounds__(256) void maxsim_fused_kernel(const float* __restrict__ pos_emb,
                                                           const float* __restrict__ neg_emb,
                                                           const int*   __restrict__ pos_mask,
                                                           const int*   __restrict__ neg_mask,
                                                           const float* __restrict__ q_n,
                                                           float* __restrict__ s_out) {
    extern __shared__ float smem[];
    float* Btile    = smem;                         // [8][16][SLAB_LD]
    float* Asm      = Btile + 8 * 16 * SLAB_LD;     // [LQ][SLAB_LD]
    float* ssq      = Asm + LQ * SLAB_LD;           // [8][DIM]
    float* invn_lds = ssq + 8 * DIM;                // [DIM]
    float* rowmax   = invn_lds + DIM;               // [8][LQ]

    const int doc  = blockIdx.x / B_;
    const int b    = blockIdx.x % B_;
    const int tid  = threadIdx.x;
    const int w    = tid >> 5;
    const int lane = tid & 31;

    const float* doc_base;
    const int*   mask_base;
    if (doc == 0) {
        doc_base  = pos_emb + (size_t)b * LD * DIM;
        mask_base = pos_mask + (size_t)b * LD;
    } else {
        const size_t nb = (size_t)(doc - 1) * B_ + b;
        doc_base  = neg_emb + nb * LD * DIM;
        mask_base = neg_mask + nb * LD;
    }
    const float* qb = q_n + (size_t)b * LQ * DIM;

    // ---- Phase 1: ssq streaming pass (RT loads -> resident in L2) --------
    // Thread (w, lane): token w + 8*it (wave-uniform -> uniform mask branch),
    // dims [4*lane, 4*lane+4) -> 512B contiguous loads per wave.
    v4f acc4 = {0.f, 0.f, 0.f, 0.f};
    #pragma unroll 4
    for (int it = 0; it < 64; ++it) {
        const int tok = w + 8 * it;
        if (mask_base[tok] != 0) {   // mask==0 rows are exact zeros: skip read
            v4f v = *(const v4f*)(doc_base + (size_t)tok * DIM + lane * 4);
            acc4 += v * v;           // mask is 0/1 -> (v*m)^2 == v^2 when m=1
        }
    }
    *(v4f*)(&ssq[w * DIM + lane * 4]) = acc4;
    __syncthreads();

    // ---- invnorm over token axis -----------------------------------------
    if (tid < DIM) {
        float ssum = 0.f;
        #pragma unroll
        for (int ww = 0; ww < 8; ++ww) ssum += ssq[ww * DIM + tid];
        invn_lds[tid] = 1.0f / fmaxf(sqrtf(ssum), EPSN);
    }
    __syncthreads();

    // ---- Stage A = q_n * invnorm -----------------------------------------
    for (int i = tid; i < LQ * DIM; i += 256) {
        const int m = i >> 7;
        const int k = i & (DIM - 1);
        Asm[m * SLAB_LD + k] = qb[i] * invn_lds[k];
    }
    __syncthreads();

    // ---- Phase 2: WMMA MaxSim GEMM ---------------------------------------
    float maxv0[8], maxv1[8];
    #pragma unroll
    for (int i = 0; i < 8; ++i) { maxv0[i] = NEG_INF; maxv1[i] = NEG_INF; }

    const int lm   = lane & 15;         // column / row-within-tile index
    const int halk = (lane >> 4) * 2;   // K sub-offset for this lane half
    float* btw = Btile + w * 16 * SLAB_LD;   // this wave's private tile

    for (int j = 0; j < 4; ++j) {
        const int t0 = (w + 8 * j) * 16;     // this wave's 16-token N-tile

        // Stage tile rows (token wave-uniform per iteration). Masked rows are
        // zero-filled without touching memory; live rows re-read through L2
        // with a non-temporal/last-use hint (no further reuse).
        #pragma unroll 4
        for (int it = 0; it < 16; ++it) {
            float* dst = btw + it * SLAB_LD + lane * 4;
            if (mask_base[t0 + it] != 0) {
                v4f v = __builtin_nontemporal_load(
                    (const v4f*)(doc_base + (size_t)(t0 + it) * DIM + lane * 4));
                *(v4f*)dst = v;
            } else {
                v4f z = {0.f, 0.f, 0.f, 0.f};
                *(v4f*)dst = z;
            }
        }
        // Tile is wave-private; same-wave LDS ordering + compiler waits suffice.

        v8f acc0 = {0.f, 0.f, 0.f, 0.f, 0.f, 0.f, 0.f, 0.f};
        v8f acc1 = {0.f, 0.f, 0.f, 0.f, 0.f, 0.f, 0.f, 0.f};

        const float* brow  = btw + lm * SLAB_LD;
        const float* arow0 = &Asm[lm * SLAB_LD];
        const float* arow1 = &Asm[(16 + lm) * SLAB_LD];

        #pragma unroll 8
        for (int ks = 0; ks < 32; ++ks) {
            const int kb = ks * 4 + halk;
            v2f a0 = { arow0[kb], arow0[kb + 1] };
            v2f a1 = { arow1[kb], arow1[kb + 1] };
            v2f bf = { brow[kb],  brow[kb + 1]  };
            acc0 = __builtin_amdgcn_wmma_f32_16x16x4_f32(false, a0, false, bf,
                                                         (short)0, acc0, false, false);
            acc1 = __builtin_amdgcn_wmma_f32_16x16x4_f32(false, a1, false, bf,
                                                         (short)0, acc1, false, false);
        }

        #pragma unroll
        for (int i = 0; i < 8; ++i) {
            maxv0[i] = fmaxf(maxv0[i], acc0[i]);
            maxv1[i] = fmaxf(maxv1[i], acc1[i]);
        }
    }

    // ---- Reductions -------------------------------------------------------
    // Max over the 16 columns held across each lane-group of 16.
    #pragma unroll
    for (int i = 0; i < 8; ++i) {
        float v0 = maxv0[i], v1 = maxv1[i];
        #pragma unroll
        for (int off = 1; off < 16; off <<= 1) {
            v0 = fmaxf(v0, __shfl_xor(v0, off, 32));
            v1 = fmaxf(v1, __shfl_xor(v1, off, 32));
        }
        maxv0[i] = v0; maxv1[i] = v1;
    }
    // Lanes 0 and 16 publish row maxes: row = mt*16 + i + 8*(lane/16).
    if ((lane & 15) == 0) {
        const int g = lane >> 4;
        #pragma unroll
        for (int i = 0; i < 8; ++i) {
            rowmax[w * LQ + i + 8 * g]      = maxv0[i];
            rowmax[w * LQ + 16 + i + 8 * g] = maxv1[i];
        }
    }
    __syncthreads();

    // Combine the 8 waves' partial row maxes, then sum 32 rows -> s[doc,b].
    if (tid < 32) {
        float m = rowmax[tid];
        #pragma unroll
        for (int ww = 1; ww < 8; ++ww) m = fmaxf(m, rowmax[ww * LQ + tid]);
        #pragma unroll
        for (int off = 1; off < 32; off <<= 1) m += __shfl_xor(m, off, 32);
        if (tid == 0) s_out[blockIdx.x] = m;
    }
}

// ---------------------------------------------------------------------------
// Kernel 3: scores[b][n] = s_neg[n,b] - s_pos[b]; log_softmax over N=8;
// KLDiv(log_target=True, batchmean). One block, thread b handles row b.
// ---------------------------------------------------------------------------
__global__ __launch_bounds__(128) void loss_kernel(const float* __restrict__ s,
                                                   const float* __restrict__ labels,
                                                   float* __restrict__ out) {
    __shared__ float red[128];
    const int b = threadIdx.x;

    const float sp = s[b];  // s[doc=0][b]
    float sc[NNEG];
    float m = NEG_INF;
    #pragma unroll
    for (int n = 0; n < NNEG; ++n) {
        sc[n] = s[(n + 1) * B_ + b] - sp;
        m = fmaxf(m, sc[n]);
    }
    float se = 0.0f;
    #pragma unroll
    for (int n = 0; n < NNEG; ++n) se += expf(sc[n] - m);
    const float lse = m + logf(se);

    float contrib = 0.0f;
    #pragma unroll
    for (int n = 0; n < NNEG; ++n) {
        const float lab = labels[(size_t)b * NNEG + n];
        contrib += expf(lab) * (lab - (sc[n] - lse));
    }
    red[b] = contrib;
    __syncthreads();
    for (int off = 64; off > 0; off >>= 1) {
        if (b < off) red[b] += red[b + off];
        __syncthreads();
    }
    if (b == 0) out[0] = red[0] / (float)B_;
}

// ---------------------------------------------------------------------------
// Launch
// ---------------------------------------------------------------------------
extern "C" void kernel_launch(void* const* d_in, const int* in_sizes, int n_in,
                              void* d_out, int out_size, void* d_ws, size_t ws_size,
                              hipStream_t stream) {
    const float* q_emb    = (const float*)d_in[0];
    const float* pos_emb  = (const float*)d_in[1];
    const float* neg_emb  = (const float*)d_in[2];
    const float* labels   = (const float*)d_in[3];
    const int*   pos_mask = (const int*)d_in[4];
    const int*   neg_mask = (const int*)d_in[5];

    float* ws  = (float*)d_ws;
    float* q_n = ws;                               // B*LQ*DIM floats
    float* s   = q_n + (size_t)B_ * LQ * DIM;      // NDOC*B floats

    qnorm_kernel<<<B_, 256, 0, stream>>>(q_emb, q_n);
    maxsim_fused_kernel<<<NDOC * B_, 256, SMEM_FLOATS * sizeof(float), stream>>>(
        pos_emb, neg_emb, pos_mask, neg_mask, q_n, s);
    loss_kernel<<<1, 128, 0, stream>>>(s, labels, (float*)d_out);
}